// LSTMPredictor_39994735460599
// MI455X (gfx1250) — compile-verified
//
#include <hip/hip_runtime.h>

#define Hh    10     // hidden size
#define G4    40     // 4*H gates
#define Bb    128    // batch
#define Tt    2048   // sequence length
#define IN2   20     // 2*H (bidirectional concat)
#define NS    24     // output steps

typedef __attribute__((ext_vector_type(2))) float v2f;
typedef __attribute__((ext_vector_type(8))) float v8f;

__device__ __forceinline__ v2f splat2(float v) { v2f r; r[0] = v; r[1] = v; return r; }

__device__ __forceinline__ float sigf(float x) {
  return __builtin_amdgcn_rcpf(1.0f + __expf(-x));   // v_exp_f32 + v_rcp_f32
}
__device__ __forceinline__ float tanhf_fast(float x) {
  float e = __expf(2.0f * x);
  return (e - 1.0f) * __builtin_amdgcn_rcpf(e + 1.0f);
}

// Gate nonlinearity + state update + packed h store, shared by both scans.
// acc[q] holds gate pre-activations j = 2q, 2q+1 (j: i[0,10) f[10,20) g[20,30) o[30,40)).
__device__ __forceinline__ void lstm_gates(const v2f acc[20], float* h, float* c,
                                           float* __restrict__ hp) {
#pragma unroll
  for (int u = 0; u < Hh; ++u) {
    const float gi = acc[u >> 1][u & 1];
    const float gf = acc[(Hh + u) >> 1][(Hh + u) & 1];
    const float gg = acc[(2 * Hh + u) >> 1][(2 * Hh + u) & 1];
    const float go = acc[(3 * Hh + u) >> 1][(3 * Hh + u) & 1];
    const float cn = fmaf(sigf(gf), c[u], sigf(gi) * tanhf_fast(gg));
    c[u] = cn;
    h[u] = sigf(go) * tanhf_fast(cn);
  }
#pragma unroll
  for (int p = 0; p < 5; ++p)
    ((float2*)hp)[p] = make_float2(h[2 * p], h[2 * p + 1]);
}

// ---------------------------------------------------------------------------
// Layer 0 scan: input size 1 (rank-1 input projection fused).
// One block per direction, one thread per batch. Whh pairs in registers,
// recurrent matmul as 200 V_PK_FMA_F32 issues/step; x load software-pipelined.
// ---------------------------------------------------------------------------
__global__ void __launch_bounds__(128)
lstm_scan0(const float* __restrict__ x,      // [B, T, 1]
           const float* __restrict__ Wih0,   // [2, 40, 1]
           const float* __restrict__ Whh0,   // [2, 40, 10]
           const float* __restrict__ b0,     // [2, 40]
           float* __restrict__ hcat)         // [T, B, 20]
{
  const int d = blockIdx.x;
  const int b = threadIdx.x;

  v2f w2[20][Hh];
#pragma unroll
  for (int q = 0; q < 20; ++q)
#pragma unroll
    for (int k = 0; k < Hh; ++k) {
      v2f t; t[0] = Whh0[(d * G4 + 2 * q) * Hh + k];
      t[1] = Whh0[(d * G4 + 2 * q + 1) * Hh + k];
      w2[q][k] = t;
    }
  v2f wx2[20], bi2[20];
#pragma unroll
  for (int q = 0; q < 20; ++q) {
    v2f a, bb;
    a[0] = Wih0[d * G4 + 2 * q];     a[1] = Wih0[d * G4 + 2 * q + 1];
    bb[0] = b0[d * G4 + 2 * q];      bb[1] = b0[d * G4 + 2 * q + 1];
    wx2[q] = a; bi2[q] = bb;
  }

  float h[Hh], c[Hh];
#pragma unroll
  for (int u = 0; u < Hh; ++u) { h[u] = 0.0f; c[u] = 0.0f; }

  const float* xb = x + (size_t)b * Tt;
  const int dt = d ? -1 : 1;
  int t = d ? (Tt - 1) : 0;
  float xv_cur = xb[t];

  for (int s = 0; s < Tt; ++s) {
    int tn = t + dt;                                  // prefetch target (clamped)
    tn = (tn < 0) ? 0 : ((tn >= Tt) ? Tt - 1 : tn);
    const float xv_next = xb[tn];                     // issue early; used next step

    v2f acc[20];
    const v2f xs = splat2(xv_cur);
#pragma unroll
    for (int q = 0; q < 20; ++q)
      acc[q] = __builtin_elementwise_fma(xs, wx2[q], bi2[q]);
#pragma unroll
    for (int k = 0; k < Hh; ++k) {
      const v2f hk = splat2(h[k]);
#pragma unroll
      for (int q = 0; q < 20; ++q)
        acc[q] = __builtin_elementwise_fma(w2[q][k], hk, acc[q]);
    }

    lstm_gates(acc, h, c, hcat + ((size_t)t * Bb + b) * IN2 + d * Hh);
    xv_cur = xv_next;
    t += dt;
  }
}

// ---------------------------------------------------------------------------
// xproj GEMM via WMMA f32 16x16x4 (unchanged math, fixed store addressing):
//   xproj[d, row, 0:40] = Xin[row, 0:20] @ Wih[d]^T + bias[d]
// ---------------------------------------------------------------------------
__global__ void __launch_bounds__(256)
xproj_gemm(const float* __restrict__ Xin,   // [T*B, 20]   (= hcat)
           const float* __restrict__ Wih,   // [2, 40, 20] (this layer)
           const float* __restrict__ bias,  // [2, 40]
           float* __restrict__ xproj)       // [2, T*B, 40]
{
  const int d = blockIdx.y;

  __shared__ float Wp[48 * 20];
  __shared__ float Bp[48];
  for (int i = threadIdx.x; i < 48 * 20; i += 256) {
    const int n = i / 20, k = i % 20;
    Wp[i] = (n < G4) ? Wih[(d * G4 + n) * 20 + k] : 0.0f;
  }
  for (int i = threadIdx.x; i < 48; i += 256)
    Bp[i] = (i < G4) ? bias[d * G4 + i] : 0.0f;
  __syncthreads();

  const int lane = threadIdx.x & 31;
  const int wave = threadIdx.x >> 5;
  const int ln16 = lane & 15;
  const int lh   = lane >> 4;

  // B tiles (W^T): vgpr v, lane l -> K = 4s + v + 2*lh, N = nt*16 + ln16.
  v2f Bt[5][3];
#pragma unroll
  for (int s = 0; s < 5; ++s)
#pragma unroll
    for (int nt = 0; nt < 3; ++nt) {
      const int n  = nt * 16 + ln16;
      const int k0 = 4 * s + 2 * lh;
      v2f bv; bv[0] = Wp[n * 20 + k0]; bv[1] = Wp[n * 20 + k0 + 1];
      Bt[s][nt] = bv;
    }
  float bv3[3];
#pragma unroll
  for (int nt = 0; nt < 3; ++nt) bv3[nt] = Bp[nt * 16 + ln16];

  const size_t obase = (size_t)d * Tt * Bb * G4;
  const int wid = blockIdx.x * 8 + wave;              // 0..1023

  for (int it = 0; it < 16; ++it) {
    const int row0 = (wid + it * 1024) * 16;          // chunk * 16
    const float* xr = Xin + (size_t)(row0 + ln16) * 20;
    if (it < 15) __builtin_prefetch(xr + 16384 * 20, 0, 1);  // next chunk's row

    v2f At[5];                                        // A: K = 4s + 2*lh + v
#pragma unroll
    for (int s = 0; s < 5; ++s)
      At[s] = *(const v2f*)(xr + 4 * s + 2 * lh);

    // One base pointer per chunk; stores use constant offsets r*160B.
    float* opb = xproj + obase + (size_t)(row0 + 8 * lh) * G4 + ln16;

#pragma unroll
    for (int nt = 0; nt < 3; ++nt) {
      v8f cacc = {};
#pragma unroll
      for (int s = 0; s < 5; ++s)
        cacc = __builtin_amdgcn_wmma_f32_16x16x4_f32(
            false, At[s], false, Bt[s][nt], (short)0, cacc, false, false);

      if (nt < 2 || ln16 < 8) {                       // n = nt*16 + ln16 < 40
        float* op = opb + nt * 16;
#pragma unroll
        for (int r = 0; r < 8; ++r)
          op[r * G4] = cacc[r] + bv3[nt];
      }
    }
  }
}

// ---------------------------------------------------------------------------
// Layers 1/2 scan step: issue next step's xproj loads, 200 packed FMAs on
// registers, then consume previously loaded xproj (latency hidden).
// ---------------------------------------------------------------------------
__device__ __forceinline__ void scan12_step(
    int t, int tn, int d, int b,
    const float* __restrict__ xproj,
    const float4* cur, float4* nxt,
    const v2f w2[20][Hh], float* h, float* c,
    float* __restrict__ hcat)
{
  const float4* gpn =
      (const float4*)(xproj + ((size_t)d * Tt * Bb + (size_t)tn * Bb + b) * G4);
#pragma unroll
  for (int q = 0; q < 10; ++q) nxt[q] = gpn[q];       // issue loads early

  v2f acc[20];
  const v2f h0 = splat2(h[0]);
#pragma unroll
  for (int q = 0; q < 20; ++q) acc[q] = w2[q][0] * h0;
#pragma unroll
  for (int k = 1; k < Hh; ++k) {
    const v2f hk = splat2(h[k]);
#pragma unroll
    for (int q = 0; q < 20; ++q)
      acc[q] = __builtin_elementwise_fma(w2[q][k], hk, acc[q]);
  }

  const float* xs = (const float*)cur;                // loaded last step
#pragma unroll
  for (int q = 0; q < 20; ++q) {
    acc[q][0] += xs[2 * q];
    acc[q][1] += xs[2 * q + 1];
  }

  lstm_gates(acc, h, c, hcat + ((size_t)t * Bb + b) * IN2 + d * Hh);
}

__global__ void __launch_bounds__(128)
lstm_scan12(const float* __restrict__ xproj, // [2, T*B, 40] (bias included)
            const float* __restrict__ Whh,   // [2, 40, 10] (this layer)
            float* __restrict__ hcat)        // [T, B, 20]
{
  const int d = blockIdx.x;
  const int b = threadIdx.x;

  v2f w2[20][Hh];
#pragma unroll
  for (int q = 0; q < 20; ++q)
#pragma unroll
    for (int k = 0; k < Hh; ++k) {
      v2f t; t[0] = Whh[(d * G4 + 2 * q) * Hh + k];
      t[1] = Whh[(d * G4 + 2 * q + 1) * Hh + k];
      w2[q][k] = t;
    }

  float h[Hh], c[Hh];
#pragma unroll
  for (int u = 0; u < Hh; ++u) { h[u] = 0.0f; c[u] = 0.0f; }

  const int dt = d ? -1 : 1;
  int t = d ? (Tt - 1) : 0;

  float4 bufA[10], bufB[10];
  const float4* gp0 =
      (const float4*)(xproj + ((size_t)d * Tt * Bb + (size_t)t * Bb + b) * G4);
#pragma unroll
  for (int q = 0; q < 10; ++q) bufA[q] = gp0[q];

  for (int s = 0; s < Tt; s += 2) {
    const int t1 = t + dt;                            // always valid (s+1 < Tt)
    scan12_step(t, t1, d, b, xproj, bufA, bufB, w2, h, c, hcat);
    int t2 = t1 + dt;
    t2 = (t2 < 0) ? 0 : ((t2 >= Tt) ? Tt - 1 : t2);   // clamp for final step
    scan12_step(t1, t2, d, b, xproj, bufB, bufA, w2, h, c, hcat);
    t += 2 * dt;
  }
}

// ---------------------------------------------------------------------------
// Final FC: out[b, s] = hcat[T-1, b, :] @ Wfc[s, :] + bfc[s]
// ---------------------------------------------------------------------------
__global__ void __launch_bounds__(128)
fc_kernel(const float* __restrict__ hcat, const float* __restrict__ Wfc,
          const float* __restrict__ bfc, float* __restrict__ out)
{
  const int b = threadIdx.x;
  float hv[IN2];
  const float* hp = hcat + ((size_t)(Tt - 1) * Bb + b) * IN2;
#pragma unroll
  for (int j = 0; j < IN2; ++j) hv[j] = hp[j];
#pragma unroll
  for (int s = 0; s < NS; ++s) {
    float acc = bfc[s];
#pragma unroll
    for (int j = 0; j < IN2; ++j) acc = fmaf(hv[j], Wfc[s * IN2 + j], acc);
    out[b * NS + s] = acc;
  }
}

// ---------------------------------------------------------------------------
extern "C" void kernel_launch(void* const* d_in, const int* in_sizes, int n_in,
                              void* d_out, int out_size, void* d_ws, size_t ws_size,
                              hipStream_t stream) {
  const float* x     = (const float*)d_in[0];  // [128, 2048, 1]
  const float* Wih0  = (const float*)d_in[1];  // [2, 40, 1]
  const float* Whh0  = (const float*)d_in[2];  // [2, 40, 10]
  const float* b0    = (const float*)d_in[3];  // [2, 40]
  const float* Wih12 = (const float*)d_in[4];  // [2, 2, 40, 20]
  const float* Whh12 = (const float*)d_in[5];  // [2, 2, 40, 10]
  const float* b12   = (const float*)d_in[6];  // [2, 2, 40]
  const float* Wfc   = (const float*)d_in[7];  // [24, 20]
  const float* bfc   = (const float*)d_in[8];  // [24]
  float* out = (float*)d_out;

  // Workspace: xproj 2*T*B*40 f32 (80 MB) + hcat T*B*20 f32 (20 MB) = 100 MB.
  float* xproj = (float*)d_ws;
  float* hcat  = xproj + (size_t)2 * Tt * Bb * G4;

  lstm_scan0<<<dim3(2), dim3(128), 0, stream>>>(x, Wih0, Whh0, b0, hcat);

  for (int l = 0; l < 2; ++l) {
    xproj_gemm<<<dim3(128, 2), dim3(256), 0, stream>>>(
        hcat, Wih12 + (size_t)l * 2 * G4 * IN2, b12 + (size_t)l * 2 * G4, xproj);
    lstm_scan12<<<dim3(2), dim3(128), 0, stream>>>(
        xproj, Whh12 + (size_t)l * 2 * G4 * Hh, hcat);
  }

  fc_kernel<<<dim3(1), dim3(128), 0, stream>>>(hcat, Wfc, bfc, out);
}